// Raster_57896159150396
// MI455X (gfx1250) — compile-verified
//
#include <hip/hip_runtime.h>
#include <math.h>

// ---------------------------------------------------------------------------
// Raster kernel for MI455X (gfx1250, wave32).
//
// Roofline: output = 131072 * 1000 fp32 = 524 MB streaming stores vs ~150
// MFLOP of math -> purely store-bandwidth bound (floor ~22.5us @ 23.3 TB/s).
// Output > 192 MB L2, so use non-temporal (NT) store hints.
//
// One depo per wave32. The 10x10 w0 (x) w1 outer product is computed with a
// single V_WMMA_F32_16X16X4_F32 (rank-1 update: only the K=0 column of A and
// K=0 row of B are non-zero), keeping full fp32 precision.
// ---------------------------------------------------------------------------

typedef __attribute__((ext_vector_type(2))) float v2f;
typedef __attribute__((ext_vector_type(8))) float v8f;

#define RW 10                         // raster window cells per dim
#define NSIGMA 3.0f
#define INV_SQRT2 0.70710678118654752440f

__global__ __launch_bounds__(256) void Raster_57896159150396_kernel(
    const float* __restrict__ sigma,    // [N,3]
    const float* __restrict__ timev,    // [N]
    const float* __restrict__ charge,   // [N]
    const float* __restrict__ tail,     // [N,3]
    const float* __restrict__ gsp,      // [3]
    float* __restrict__ rasters,        // [N,10,10,10]
    float* __restrict__ offsets,        // [N,3] (int values stored as f32)
    int N)
{
  const int lane = threadIdx.x & 31;
  const int n    = blockIdx.x * 8 + (threadIdx.x >> 5);   // one depo per wave
  if (n >= N) return;                                     // uniform per wave

  // --- per-depo scalars (broadcast loads; uniform across the wave) ---------
  const float g0 = gsp[0], g1 = gsp[1], g2 = gsp[2];
  const float s0 = sigma[3 * n + 0];
  const float s1 = sigma[3 * n + 1];
  const float s2 = sigma[3 * n + 2];
  const float p0 = tail[3 * n + 1];                       // pdims=(1,2)
  const float p1 = tail[3 * n + 2];
  const float p2 = timev[n];                              // tdim=-1
  const float q  = charge[n];

  const float im0 = floorf((p0 - NSIGMA * s0) / g0);
  const float im1 = floorf((p1 - NSIGMA * s1) / g1);
  const float im2 = floorf((p2 - NSIGMA * s2) / g2);

  // --- lane k = edge k: CDF for all 3 dims (lanes 11..31 compute garbage
  //     that saturates harmlessly; keeps EXEC full, no divergence) ----------
  const float k  = (float)lane;
  const float c0 = 0.5f * (1.0f + erff(((im0 + k) * g0 - p0) * INV_SQRT2 / s0));
  const float c1 = 0.5f * (1.0f + erff(((im1 + k) * g1 - p1) * INV_SQRT2 / s1));
  const float c2 = 0.5f * (1.0f + erff(((im2 + k) * g2 - p2) * INV_SQRT2 / s2));

  // per-dim cell weights, valid in lanes 0..9
  const float w0 = __shfl_down(c0, 1) - c0;
  const float w1 = __shfl_down(c1, 1) - c1;
  const float w2 = __shfl_down(c2, 1) - c2;

  // --- rank-1 outer product via WMMA f32 16x16x4 ---------------------------
  // A (16x4): lanes 0..15 / VGPR0 hold K=0 column -> put w0 there, zero rest.
  // B (4x16): lanes 0..15 / VGPR0 hold K=0 row    -> put w1 there, zero rest.
  v2f a, b;
  a.x = (lane < RW) ? w0 : 0.0f;  a.y = 0.0f;
  b.x = (lane < RW) ? w1 : 0.0f;  b.y = 0.0f;
  v8f acc = {};
  acc = __builtin_amdgcn_wmma_f32_16x16x4_f32(
      /*neg_a=*/false, a, /*neg_b=*/false, b,
      /*c_mod=*/(short)0, acc, /*reuse_a=*/false, /*reuse_b=*/false);
  // D layout: row = v + 8*(lane>=16), col = lane & 15.

  // --- broadcast charge-scaled z-weights to all lanes ----------------------
  float qs[RW];
#pragma unroll
  for (int z = 0; z < RW; ++z) qs[z] = q * __shfl(w2, z);

  // --- stream the 10x10x10 tile with non-temporal b64 stores ---------------
  const int col     = lane & 15;
  const int rowbase = (lane >> 4) * 8;
  float* __restrict__ outn = rasters + (size_t)n * (RW * RW * RW);
#pragma unroll
  for (int v = 0; v < 8; ++v) {
    const int row = rowbase + v;
    if (row < RW && col < RW) {
      const float wxy = acc[v];
      float* p = outn + row * (RW * RW) + col * RW;       // 8B-aligned
#pragma unroll
      for (int z = 0; z < RW; z += 2) {
        v2f val;
        val.x = wxy * qs[z];
        val.y = wxy * qs[z + 1];
        __builtin_nontemporal_store(val, (v2f*)(p + z));  // global_store_b64 NT
      }
    }
  }

  // --- offsets: int32 grid indices, emitted as f32 values into the f32
  //     concatenated output buffer ------------------------------------------
  if (lane < 3) {
    const float v = (lane == 0) ? im0 : (lane == 1) ? im1 : im2;
    offsets[(size_t)n * 3 + lane] = v;
  }
}

extern "C" void kernel_launch(void* const* d_in, const int* in_sizes, int n_in,
                              void* d_out, int out_size, void* d_ws, size_t ws_size,
                              hipStream_t stream) {
  (void)n_in; (void)out_size; (void)d_ws; (void)ws_size;

  const float* sigma  = (const float*)d_in[0];   // [N,3]
  const float* timev  = (const float*)d_in[1];   // [N]
  const float* charge = (const float*)d_in[2];   // [N]
  const float* tail   = (const float*)d_in[3];   // [N,3]
  const float* gsp    = (const float*)d_in[4];   // [3]

  const int N = in_sizes[1];                     // time is [N]

  float* rasters = (float*)d_out;                        // N*1000 floats
  float* offsets = (float*)d_out + (size_t)N * 1000;     // N*3 values

  const int wavesPerBlock = 8;                   // 256 threads = 8 wave32
  dim3 block(256);
  dim3 grid((N + wavesPerBlock - 1) / wavesPerBlock);
  Raster_57896159150396_kernel<<<grid, block, 0, stream>>>(
      sigma, timev, charge, tail, gsp, rasters, offsets, N);
}